// SupContrastiveLoss_53704271069699
// MI455X (gfx1250) — compile-verified
//
#include <hip/hip_runtime.h>
#include <math.h>
#include <stdint.h>

typedef float v2f __attribute__((ext_vector_type(2)));
typedef float v8f __attribute__((ext_vector_type(8)));

#define B_ROWS 8192
#define D_DIM  1024
#define N_COLS 16384
#define NBLK   128          // column blocks of 128 -> partials per row
#define INV_TEMP 20.0f
#define EPS_NORM 1e-8f

// ---------------------------------------------------------------------------
// CDNA5 async global->LDS copy (16B per lane), tracked by ASYNCcnt.
// LDS aperture: the LDS address is the low 32 bits of the flat one, so truncating the generic
// pointer gives the LDS byte offset the instruction expects in its VGPR.
// ---------------------------------------------------------------------------
__device__ __forceinline__ void async_b128_to_lds(void* lds_dst,
                                                  const float* gsrc) {
  uint32_t laddr = (uint32_t)(uintptr_t)lds_dst;
  asm volatile("global_load_async_to_lds_b128 %0, %1, off"
               :
               : "v"(laddr), "v"(gsrc)
               : "memory");
}

__device__ __forceinline__ void wait_async_all() {
  asm volatile("s_wait_asynccnt 0x0" ::: "memory");
}

// ---------------------------------------------------------------------------
// Kernel 1: inverse L2 row norms for premise/entail/contra.
// grid (8192, 3), block 256. Each thread loads one float4 (4*256 = 1024 = D).
// ---------------------------------------------------------------------------
__global__ __launch_bounds__(256) void scl_rownorm_kernel(
    const float* __restrict__ p, const float* __restrict__ e,
    const float* __restrict__ c, float* __restrict__ inv) {
  __shared__ float red[256];
  const float* src = (blockIdx.y == 0) ? p : (blockIdx.y == 1) ? e : c;
  const int row = blockIdx.x;
  const float4 v =
      *(const float4*)(src + (size_t)row * D_DIM + (size_t)threadIdx.x * 4);
  red[threadIdx.x] = v.x * v.x + v.y * v.y + v.z * v.z + v.w * v.w;
  __syncthreads();
  for (int off = 128; off > 0; off >>= 1) {
    if (threadIdx.x < (unsigned)off) red[threadIdx.x] += red[threadIdx.x + off];
    __syncthreads();
  }
  if (threadIdx.x == 0) {
    float n = sqrtf(red[0]);
    inv[(size_t)blockIdx.y * B_ROWS + row] = 1.0f / fmaxf(n, EPS_NORM);
  }
}

// ---------------------------------------------------------------------------
// Kernel 2: WMMA fp32 GEMM -> sims[8192][16384] (scaled), PLUS fused
// per-tile partial logsumexp so sims is never re-read.
// grid (16384/128, 8192/128), block 256.  blockIdx.x < 64 -> B=entail else
// contra.  Double-buffered LDS, async global->LDS staging overlapped with
// WMMA.
// ---------------------------------------------------------------------------
__global__ __launch_bounds__(256) void scl_gemm_wmma_kernel(
    const float* __restrict__ p, const float* __restrict__ e,
    const float* __restrict__ c, const float* __restrict__ invAll,
    float* __restrict__ sims, float* __restrict__ pmaxWS,
    float* __restrict__ psumWS) {
  // Row-major padded tiles for BOTH operands: fragment reads are adjacent
  // k-pairs (ds_load_b64) at lane-stride 20 floats -> 16 distinct banks;
  // 20 % 4 == 0 keeps every 16B chunk aligned for the async B128 copies.
  __shared__ float As[2][128][20];  // [buf][m][k]
  __shared__ float Bs[2][128][20];  // [buf][n][k]
  __shared__ float pm[128][2];      // per-row tile max, per N-wave
  __shared__ float ps[128][2];      // per-row tile sumexp, per N-wave

  const int tid   = threadIdx.x;
  const int lane  = tid & 31;
  const int wave  = tid >> 5;     // 0..7
  const int waveM = wave & 3;     // 4 waves along M -> 32 rows each
  const int waveN = wave >> 2;    // 2 waves along N -> 64 cols each
  const int r     = lane & 15;    // row (A) / col (B) within a 16-tile
  const int hk    = lane >> 4;    // K-half selector per ISA fragment layout

  const int rowBase = blockIdx.y * 128;
  const int CN      = blockIdx.x * 128;          // global sims column base
  const bool firstHalf = (CN < B_ROWS);
  const float* __restrict__ Bsrc = firstHalf ? e : c;
  const float* __restrict__ invP = invAll;
  const float* __restrict__ invB = firstHalf ? (invAll + B_ROWS)
                                             : (invAll + 2 * B_ROWS);
  const int cnLocal = CN & (B_ROWS - 1);

  // Per-thread staging assignment: 2 x (row, 16B chunk) per matrix.
  const int srow0 = tid >> 2;            // 0..63
  const int srow1 = (tid + 256) >> 2;    // 64..127
  const int skq   = (tid & 3) << 2;      // 0,4,8,12

  auto stage = [&](int buf, int kt) {
    async_b128_to_lds(&As[buf][srow0][skq],
                      p + (size_t)(rowBase + srow0) * D_DIM + kt + skq);
    async_b128_to_lds(&Bs[buf][srow0][skq],
                      Bsrc + (size_t)(cnLocal + srow0) * D_DIM + kt + skq);
    async_b128_to_lds(&As[buf][srow1][skq],
                      p + (size_t)(rowBase + srow1) * D_DIM + kt + skq);
    async_b128_to_lds(&Bs[buf][srow1][skq],
                      Bsrc + (size_t)(cnLocal + srow1) * D_DIM + kt + skq);
  };

  v8f acc[2][4] = {};

  stage(0, 0);
  wait_async_all();
  __syncthreads();

  for (int kt = 0; kt < D_DIM; kt += 16) {
    const int cur = (kt >> 4) & 1;
    if (kt + 16 < D_DIM) stage(cur ^ 1, kt + 16);  // overlap DMA with WMMA

#pragma unroll
    for (int k0 = 0; k0 < 16; k0 += 4) {
      v2f af[2];
      v2f bf[4];
#pragma unroll
      for (int mt = 0; mt < 2; ++mt) {
        const int mrow = waveM * 32 + mt * 16 + r;
        af[mt].x = As[cur][mrow][k0 + 2 * hk];
        af[mt].y = As[cur][mrow][k0 + 2 * hk + 1];
      }
#pragma unroll
      for (int nt = 0; nt < 4; ++nt) {
        const int ncol = waveN * 64 + nt * 16 + r;
        bf[nt].x = Bs[cur][ncol][k0 + 2 * hk];
        bf[nt].y = Bs[cur][ncol][k0 + 2 * hk + 1];
      }
#pragma unroll
      for (int mt = 0; mt < 2; ++mt)
#pragma unroll
        for (int nt = 0; nt < 4; ++nt)
          acc[mt][nt] = __builtin_amdgcn_wmma_f32_16x16x4_f32(
              false, af[mt], false, bf[nt], (short)0, acc[mt][nt], false,
              false);
    }

    wait_async_all();
    __syncthreads();
  }

  // -------------------------------------------------------------------------
  // Epilogue: scale by invP[row]*invB[col]*(1/TEMP), store sims (scalar 4B
  // stores: sims = d_out+1), and build per-row partial logsumexp.
  // Lanes {0..15} / {16..31} of a wave hold rows i / i+8 of a 16-tile (columns in
  // r), so width-16 shuffles reduce across the tile's 64 columns per wave.
  // -------------------------------------------------------------------------
  float sbv[4];
#pragma unroll
  for (int nt = 0; nt < 4; ++nt)
    sbv[nt] = invB[cnLocal + waveN * 64 + nt * 16 + r] * INV_TEMP;

#pragma unroll
  for (int mt = 0; mt < 2; ++mt) {
#pragma unroll
    for (int i = 0; i < 8; ++i) {
      const int rowL = waveM * 32 + mt * 16 + i + 8 * hk;
      const int grow = rowBase + rowL;
      const float ip = invP[grow];
      float v4[4];
      float m = -INFINITY;
#pragma unroll
      for (int nt = 0; nt < 4; ++nt) {
        v4[nt] = acc[mt][nt][i] * ip * sbv[nt];
        m = fmaxf(m, v4[nt]);
      }
#pragma unroll
      for (int nt = 0; nt < 4; ++nt)
        sims[(size_t)grow * N_COLS + CN + waveN * 64 + nt * 16 + r] = v4[nt];
      // row max across the 64 columns held by this shuffle group
#pragma unroll
      for (int off = 8; off > 0; off >>= 1)
        m = fmaxf(m, __shfl_xor(m, off, 16));
      float s = 0.0f;
#pragma unroll
      for (int nt = 0; nt < 4; ++nt) s += expf(v4[nt] - m);
#pragma unroll
      for (int off = 8; off > 0; off >>= 1) s += __shfl_xor(s, off, 16);
      if (r == 0) {  // one writer per shuffle group; both hk halves write
        pm[rowL][waveN] = m;
        ps[rowL][waveN] = s;
      }
    }
  }
  __syncthreads();

  if (tid < 128) {
    const float m0 = pm[tid][0], m1 = pm[tid][1];
    const float M = fmaxf(m0, m1);
    const float S = ps[tid][0] * expf(m0 - M) + ps[tid][1] * expf(m1 - M);
    const int grow = rowBase + tid;
    pmaxWS[(size_t)grow * NBLK + blockIdx.x] = M;
    psumWS[(size_t)grow * NBLK + blockIdx.x] = S;
  }
}

// ---------------------------------------------------------------------------
// Kernel 3: merge 128 per-row partials -> rowloss (no sims re-read except
// the diagonal element).  grid 8192, block 128.
// ---------------------------------------------------------------------------
__global__ __launch_bounds__(128) void scl_lse_combine_kernel(
    const float* __restrict__ pmaxWS, const float* __restrict__ psumWS,
    const float* __restrict__ sims, float* __restrict__ rowloss) {
  __shared__ float rm[128];
  __shared__ float rs[128];
  const int row = blockIdx.x;
  const int tid = threadIdx.x;
  const float m = pmaxWS[(size_t)row * NBLK + tid];
  const float s = psumWS[(size_t)row * NBLK + tid];

  rm[tid] = m;
  __syncthreads();
  for (int off = 64; off > 0; off >>= 1) {
    if (tid < off) rm[tid] = fmaxf(rm[tid], rm[tid + off]);
    __syncthreads();
  }
  const float M = rm[0];
  __syncthreads();

  rs[tid] = s * expf(m - M);
  __syncthreads();
  for (int off = 64; off > 0; off >>= 1) {
    if (tid < off) rs[tid] += rs[tid + off];
    __syncthreads();
  }
  if (tid == 0)
    rowloss[row] = M + logf(rs[0]) - sims[(size_t)row * N_COLS + row];
}

// ---------------------------------------------------------------------------
// Kernel 4: mean of 8192 per-row losses -> d_out[0]. 1 block of 256.
// ---------------------------------------------------------------------------
__global__ __launch_bounds__(256) void scl_mean_kernel(
    const float* __restrict__ rowloss, float* __restrict__ out) {
  __shared__ float red[256];
  float s = 0.0f;
  for (int j = threadIdx.x; j < B_ROWS; j += 256) s += rowloss[j];
  red[threadIdx.x] = s;
  __syncthreads();
  for (int off = 128; off > 0; off >>= 1) {
    if (threadIdx.x < (unsigned)off) red[threadIdx.x] += red[threadIdx.x + off];
    __syncthreads();
  }
  if (threadIdx.x == 0) out[0] = red[0] / (float)B_ROWS;
}

extern "C" void kernel_launch(void* const* d_in, const int* in_sizes, int n_in,
                              void* d_out, int out_size, void* d_ws,
                              size_t ws_size, hipStream_t stream) {
  (void)in_sizes; (void)n_in; (void)out_size; (void)ws_size;
  const float* p = (const float*)d_in[0];
  const float* e = (const float*)d_in[1];
  const float* c = (const float*)d_in[2];
  float* out  = (float*)d_out;
  float* sims = out + 1;                         // output order: (loss, sims)
  float* inv     = (float*)d_ws;                 // 3 * 8192 floats
  float* rowloss = inv + 3 * B_ROWS;             // 8192 floats
  float* pmaxWS  = rowloss + B_ROWS;             // 8192 * 128 floats (4 MB)
  float* psumWS  = pmaxWS + (size_t)B_ROWS * NBLK;  // 8192 * 128 floats

  scl_rownorm_kernel<<<dim3(B_ROWS, 3), 256, 0, stream>>>(p, e, c, inv);
  scl_gemm_wmma_kernel<<<dim3(N_COLS / 128, B_ROWS / 128), 256, 0, stream>>>(
      p, e, c, inv, sims, pmaxWS, psumWS);
  scl_lse_combine_kernel<<<B_ROWS, 128, 0, stream>>>(pmaxWS, psumWS, sims,
                                                     rowloss);
  scl_mean_kernel<<<1, 256, 0, stream>>>(rowloss, out);
}